// XMaskSwitchTop2Hard_78606491452075
// MI455X (gfx1250) — compile-verified
//
#include <hip/hip_runtime.h>

// XMaskSwitchTop2Hard — forward pass.
// out[i] = x[i]                    if |x[i]| in top-2 abs of its group-of-4
//        = sigmoid(logit_g)*x[i]   otherwise        (g = group index % 1024)
//
// Memory-bound streaming kernel: clause-batched 128-bit NT loads/stores,
// sigmoid table amortized in LDS, v_rcp-based sigmoid.

typedef float v4f __attribute__((ext_vector_type(4)));

#define GROUPS_PER_ROW 1024   // 4096 / GROUP(4), matches gate_logits size
#define BLOCK          256    // 8 wave32 waves
#define ROWS_PER_BLOCK 4      // groups per block = 4096
#define GROUPS_PER_BLOCK (GROUPS_PER_ROW * ROWS_PER_BLOCK)
#define ITEMS (GROUPS_PER_BLOCK / BLOCK)   // 16 float4 per thread

__device__ __forceinline__ float fast_sigmoid(float z)
{
    // v_mul + v_exp_f32 + v_add + v_rcp_f32  (~1 ulp, ample for a blend weight)
    return __builtin_amdgcn_rcpf(1.0f + __expf(-z));
}

__device__ __forceinline__ v4f top2_mix(v4f v, float r)
{
    const float s0 = fabsf(v.x), s1 = fabsf(v.y);
    const float s2 = fabsf(v.z), s3 = fabsf(v.w);

    // c_i = #elements beating element i under top_k's (value, lower-index)
    // ordering. Element kept unchanged iff c_i < 2.
    const int c0 = (s1 >  s0) + (s2 >  s0) + (s3 >  s0);
    const int c1 = (s0 >= s1) + (s2 >  s1) + (s3 >  s1);
    const int c2 = (s0 >= s2) + (s1 >= s2) + (s3 >  s2);
    const int c3 = (s0 >= s3) + (s1 >= s3) + (s2 >= s3);

    v4f o;
    o.x = (c0 < 2) ? v.x : r * v.x;
    o.y = (c1 < 2) ? v.y : r * v.y;
    o.z = (c2 < 2) ? v.z : r * v.z;
    o.w = (c3 < 2) ? v.w : r * v.w;
    return o;
}

// Main kernel: every block processes exactly GROUPS_PER_BLOCK groups —
// no bounds checks, so all 16 NT b128 loads can issue as one clause.
__global__ __launch_bounds__(BLOCK) void
XMaskSwitchTop2Hard_78606491452075_kernel(const float* __restrict__ x,
                                          const float* __restrict__ gate_logits,
                                          float* __restrict__ out)
{
    __shared__ float r_lds[GROUPS_PER_ROW];

    const int t = threadIdx.x;

    // Build per-row sigmoid table once per block (amortized over 4 rows).
    #pragma unroll
    for (int i = 0; i < GROUPS_PER_ROW / BLOCK; ++i) {
        const int g = i * BLOCK + t;
        r_lds[g] = fast_sigmoid(gate_logits[g]);
    }
    __syncthreads();

    const long long base = (long long)blockIdx.x * GROUPS_PER_BLOCK;
    const v4f* __restrict__ xv = (const v4f*)x;
    v4f*       __restrict__ ov = (v4f*)out;

    v4f vbuf[ITEMS];

    // Batch all streaming reads first (touched once -> NT, keep L2 clean).
    #pragma unroll
    for (int k = 0; k < ITEMS; ++k)
        vbuf[k] = __builtin_nontemporal_load(&xv[base + (long long)k * BLOCK + t]);

    #pragma unroll
    for (int k = 0; k < ITEMS; ++k) {
        // base is a multiple of GROUPS_PER_ROW, so in-row group index:
        const float r = r_lds[(k * BLOCK + t) & (GROUPS_PER_ROW - 1)];
        const v4f o = top2_mix(vbuf[k], r);
        __builtin_nontemporal_store(o, &ov[base + (long long)k * BLOCK + t]);
    }
}

// Tail kernel: one guarded group per thread. Launches 0 blocks for the
// reference shape; exists only to keep kernel_launch shape-agnostic.
__global__ __launch_bounds__(BLOCK) void
XMaskSwitchTop2Hard_78606491452075_tail(const float* __restrict__ x,
                                        const float* __restrict__ gate_logits,
                                        float* __restrict__ out,
                                        long long first_group,
                                        long long n_groups)
{
    const long long gi = first_group
                       + (long long)blockIdx.x * BLOCK + threadIdx.x;
    if (gi >= n_groups) return;

    const v4f* __restrict__ xv = (const v4f*)x;
    v4f*       __restrict__ ov = (v4f*)out;

    const int g = (int)(gi & (GROUPS_PER_ROW - 1));
    const float r = fast_sigmoid(gate_logits[g]);

    const v4f v = __builtin_nontemporal_load(&xv[gi]);
    __builtin_nontemporal_store(top2_mix(v, r), &ov[gi]);
}

extern "C" void kernel_launch(void* const* d_in, const int* in_sizes, int n_in,
                              void* d_out, int out_size, void* d_ws, size_t ws_size,
                              hipStream_t stream)
{
    (void)n_in; (void)out_size; (void)d_ws; (void)ws_size;

    const float* x           = (const float*)d_in[0];
    const float* gate_logits = (const float*)d_in[1];
    float*       out         = (float*)d_out;

    const long long n_elems  = (long long)in_sizes[0];   // 4*8192*4096
    const long long n_groups = n_elems / 4;              // 33,554,432

    const long long full_blocks = n_groups / GROUPS_PER_BLOCK;   // 8192
    const long long done        = full_blocks * GROUPS_PER_BLOCK;
    const long long rem         = n_groups - done;               // 0 for ref

    if (full_blocks > 0) {
        XMaskSwitchTop2Hard_78606491452075_kernel
            <<<dim3((unsigned)full_blocks), dim3(BLOCK), 0, stream>>>(
                x, gate_logits, out);
    }
    if (rem > 0) {
        const unsigned tail_blocks = (unsigned)((rem + BLOCK - 1) / BLOCK);
        XMaskSwitchTop2Hard_78606491452075_tail
            <<<dim3(tail_blocks), dim3(BLOCK), 0, stream>>>(
                x, gate_logits, out, done, n_groups);
    }
}